// ConvWithFilter_45079976739070
// MI455X (gfx1250) — compile-verified
//
#include <hip/hip_runtime.h>
#include <hip/hip_bf16.h>

typedef __attribute__((ext_vector_type(2))) float v2f;
typedef __attribute__((ext_vector_type(8))) float v8f;

// Problem constants from the reference: N=2, C=32, H=W=64, K=3, O=32
#define CH    32
#define HW    64
#define LLOC  (HW * HW)      // 4096 locations per batch
#define CKK   (CH * 9)       // 288
#define OCH   32

// One wave32 per output location (n,l).
//   A (16x4 f32)  = filters^T tile: A[m][k] = W[kk+k][o0+m]
//   B (4x16 f32)  = feature chunk broadcast across all 16 columns
//   D (16x16 f32) = output replicated across columns; take column per half-wave.
// A VGPR layout (ISA 7.12.2, 32-bit 16x4): V0 = {K=0 | K=2}, V1 = {K=1 | K=3}
// (lanes 0-15 | lanes 16-31), lane%16 = M.  B assumed mirrored: V0 = {K=0 | K=2} rows.
__global__ __launch_bounds__(128)
void conv_with_filter_wmma(const float* __restrict__ features,
                           const float* __restrict__ filters,
                           float* __restrict__ out,
                           int total_loc) {
    __shared__ float sfeat[4][CKK];

    const int lane = threadIdx.x & 31;
    const int wid  = threadIdx.x >> 5;
    const int idx  = blockIdx.x * 4 + wid;          // global location id: n*4096 + l
    if (idx >= total_loc) return;                    // uniform per wave (grid sized exactly)

    const int n = idx >> 12;          // / 4096
    const int l = idx & (LLOC - 1);
    const int h = l >> 6;
    const int w = l & (HW - 1);

    // ---- Stage unfolded feature vector in LDS: lane = channel, 9 window taps each.
    {
        const size_t chbase = ((size_t)n * CH + lane) * (size_t)(HW * HW);
        #pragma unroll
        for (int kh = 0; kh < 3; ++kh) {
            #pragma unroll
            for (int kw = 0; kw < 3; ++kw) {
                const int hh = h + kh - 1;
                const int ww = w + kw - 1;
                float v = 0.0f;
                if (hh >= 0 && hh < HW && ww >= 0 && ww < HW)
                    v = features[chbase + (size_t)hh * HW + ww];
                sfeat[wid][lane * 9 + kh * 3 + kw] = v;
            }
        }
    }
    __syncthreads();

    // ---- Per-location filter block: contiguous 288x32 f32 (36 KB), streamed once.
    const float* __restrict__ fb = filters + (size_t)idx * (size_t)(CKK * OCH);
    const int o   = lane & 15;            // M within tile
    const int khi = (lane >> 4) << 1;     // 0 for lanes 0-15, 2 for lanes 16-31

    v8f acc0 = {};   // output channels  0..15
    v8f acc1 = {};   // output channels 16..31

    for (int kk = 0; kk < CKK; kk += 4) {
        // B: feature broadcast (LDS, two distinct addresses per wave -> broadcast reads)
        v2f B;
        B.x = sfeat[wid][kk + khi + 0];
        B.y = sfeat[wid][kk + khi + 1];

        // A: filters^T gathers, non-temporal (302 MB single-use stream; keep L2 clean)
        const float* fk = fb + (size_t)(kk + khi) * OCH + o;
        v2f A0, A1;
        A0.x = __builtin_nontemporal_load(fk);
        A0.y = __builtin_nontemporal_load(fk + OCH);
        A1.x = __builtin_nontemporal_load(fk + 16);
        A1.y = __builtin_nontemporal_load(fk + OCH + 16);

        acc0 = __builtin_amdgcn_wmma_f32_16x16x4_f32(false, A0, false, B,
                                                     (short)0, acc0, false, false);
        acc1 = __builtin_amdgcn_wmma_f32_16x16x4_f32(false, A1, false, B,
                                                     (short)0, acc1, false, false);
    }

    // ---- D[m][*] replicated across columns: VGPR r = {M=r | M=8+r}.
    // Lanes 0 and 16 write out 32 ReLU'd channels: out[n][o][l], stride L across o.
    if ((lane & 15) == 0) {
        const int mb = (lane >> 4) << 3;                  // 0 or 8
        float* ob = out + (size_t)n * (OCH * LLOC) + l;
        #pragma unroll
        for (int r = 0; r < 8; ++r) {
            ob[(size_t)(mb + r) * LLOC]      = fmaxf(acc0[r], 0.0f);
            ob[(size_t)(16 + mb + r) * LLOC] = fmaxf(acc1[r], 0.0f);
        }
    }
}

extern "C" void kernel_launch(void* const* d_in, const int* in_sizes, int n_in,
                              void* d_out, int out_size, void* d_ws, size_t ws_size,
                              hipStream_t stream) {
    const float* features = (const float*)d_in[0];   // [N, 32, 64, 64] f32
    const float* filters  = (const float*)d_in[1];   // [N, 4096, 288, 32] f32
    float*       out      = (float*)d_out;           // [N, 32, 64, 64] f32

    const int total_loc = in_sizes[1] / (CKK * OCH); // N * 4096 = 8192
    const int waves_per_block = 4;                   // 128 threads, wave32
    const int grid = (total_loc + waves_per_block - 1) / waves_per_block;

    conv_with_filter_wmma<<<grid, 128, 0, stream>>>(features, filters, out, total_loc);
}